// FullGINModel_49976239456904
// MI455X (gfx1250) — compile-verified
//
#include <hip/hip_runtime.h>
#include <math.h>

// Problem constants (match reference)
#define NCLI 100000
#define NSKU_N 50000
#define NEDGE 600000
#define CH 128
#define FEAT 64
#define NCAT_O 100
#define NSKU_O 2000

// LDS staging geometry for the WMMA GEMM
#define KCHUNK 64
#define NBLK 128           // cols per block (8 waves x 16)
#define BS_STRIDE 144      // (NBLK+16) dwords: half-wave K rows land 32 banks apart
#define AS_STRIDE 68       // (KCHUNK+4) dwords: conflict-free b64 fragment reads

typedef __attribute__((ext_vector_type(2))) float v2f;
typedef __attribute__((ext_vector_type(8))) float v8f;

__device__ __forceinline__ float wave_sum(float v) {
#pragma unroll
  for (int off = 16; off > 0; off >>= 1) v += __shfl_xor(v, off, 32);
  return v;
}

// ---------------- per-row 1/max(||x||,1e-12) over F cols ----------------
__global__ void rownorm_kernel(const float* __restrict__ x, float* __restrict__ invn,
                               int rows, int F) {
  int wid = (int)((blockIdx.x * blockDim.x + threadIdx.x) >> 5);
  int lane = threadIdx.x & 31;
  if (wid >= rows) return;
  const float* row = x + (size_t)wid * F;
  float s = 0.f;
  for (int c = lane; c < F; c += 32) { float v = row[c]; s += v * v; }
  s = wave_sum(s);
  if (lane == 0) invn[wid] = 1.0f / fmaxf(sqrtf(s), 1e-12f);
}

// ---------------- WMMA f32 GEMM: C = act(rowscale .* A @ W + b) ----------------
// A: [M,K] row-major, W: [K,N] row-major, C: [M,N] row-major.
// Block = 8 waves: one 16-row M-tile x 128-col N-block; wave w owns N-tile w.
// A/B staged through LDS (coalesced global loads, conflict-free ds fragment reads).
// K must be a multiple of KCHUNK (64 or 128 here). ACT: 0=none 1=relu 2=sigmoid
template <int ACT>
__global__ __launch_bounds__(256) void gemm_wmma_kernel(
    const float* __restrict__ A, const float* __restrict__ rowscale,
    const float* __restrict__ W, const float* __restrict__ bias,
    float* __restrict__ C, int M, int N, int K) {
  __shared__ float As[16 * AS_STRIDE];
  __shared__ float Bs[KCHUNK * BS_STRIDE];

  const int tid = threadIdx.x;
  const int lane = tid & 31;
  const int wave = tid >> 5;
  const int r = lane & 15;      // row (A frag) / col-in-tile (B frag)
  const int half = lane >> 4;   // selects K pair {0,1} vs {2,3}
  const int kA = 2 * half;
  const int m0 = blockIdx.x * 16;
  const int ncol0 = blockIdx.y * NBLK;
  const int cl = wave * 16 + r;          // column within the 128-col block
  const int col = ncol0 + cl;            // global output column

  v8f acc0 = (v8f){0.f, 0.f, 0.f, 0.f, 0.f, 0.f, 0.f, 0.f};
  v8f acc1 = acc0;

  for (int kc = 0; kc < K; kc += KCHUNK) {
    // ---- stage A chunk: 16 x KCHUNK, coalesced along K ----
#pragma unroll
    for (int i = tid; i < 16 * KCHUNK; i += 256) {
      int row = i >> 6, c = i & (KCHUNK - 1);
      As[row * AS_STRIDE + c] = A[(size_t)(m0 + row) * K + kc + c];
    }
    // ---- stage B chunk: KCHUNK x NBLK, coalesced along N, zero-padded ----
#pragma unroll
    for (int i = tid; i < KCHUNK * NBLK; i += 256) {
      int row = i >> 7, c = i & (NBLK - 1);
      int gcol = ncol0 + c;
      Bs[row * BS_STRIDE + c] =
          (gcol < N) ? W[(size_t)(kc + row) * N + gcol] : 0.f;
    }
    __syncthreads();

    if (kc + KCHUNK < K)  // warm next B chunk while we compute
      __builtin_prefetch(&W[(size_t)(kc + KCHUNK) * N + ncol0 + lane], 0, 1);

    // ---- 16 k-steps, two interleaved accumulation chains for ILP ----
#pragma unroll
    for (int kb = 0; kb < KCHUNK; kb += 8) {
      const int k0 = kb + kA;
      const int k1 = kb + 4 + kA;
      v2f a0 = *(const v2f*)&As[r * AS_STRIDE + k0];
      v2f a1 = *(const v2f*)&As[r * AS_STRIDE + k1];
      v2f b0, b1;
      b0.x = Bs[k0 * BS_STRIDE + cl];
      b0.y = Bs[(k0 + 1) * BS_STRIDE + cl];
      b1.x = Bs[k1 * BS_STRIDE + cl];
      b1.y = Bs[(k1 + 1) * BS_STRIDE + cl];
      acc0 = __builtin_amdgcn_wmma_f32_16x16x4_f32(false, a0, false, b0,
                                                   (short)0, acc0, false, false);
      acc1 = __builtin_amdgcn_wmma_f32_16x16x4_f32(false, a1, false, b1,
                                                   (short)0, acc1, false, false);
    }
    __syncthreads();
  }

  if (col < N) {
    float bv = bias[col];
#pragma unroll
    for (int v = 0; v < 8; ++v) {
      int rowm = m0 + v + 8 * half;
      float val = acc0[v] + acc1[v];
      if (rowscale) val *= rowscale[rowm];
      val += bv;
      if (ACT == 1) val = fmaxf(val, 0.f);
      else if (ACT == 2) val = 1.0f / (1.0f + __expf(-val));
      C[(size_t)rowm * N + col] = val;
    }
  }
}

// ---------------- agg init: y = (1+eps)*x ----------------
__global__ void scale_copy_kernel(const float* __restrict__ x, float* __restrict__ y,
                                  const float* __restrict__ eps, size_t n) {
  float s = 1.0f + eps[0];
  size_t stride = (size_t)gridDim.x * blockDim.x;
  for (size_t i = (size_t)blockIdx.x * blockDim.x + threadIdx.x; i < n; i += stride)
    y[i] = s * x[i];
}

// ---------------- edge scatter-add: agg[dst] += xsrc[src], one wave per edge ----------------
__global__ void scatter_add_kernel(float* __restrict__ agg, const float* __restrict__ xsrc,
                                   const int* __restrict__ src, const int* __restrict__ dst,
                                   int E) {
  int e = (int)((blockIdx.x * blockDim.x + threadIdx.x) >> 5);
  int lane = threadIdx.x & 31;
  if (e >= E) return;
  int s = src[e], d = dst[e];
  const float4 v = ((const float4*)(xsrc + (size_t)s * CH))[lane];
  float* dp = agg + (size_t)d * CH + lane * 4;
  atomicAdd(dp + 0, v.x);
  atomicAdd(dp + 1, v.y);
  atomicAdd(dp + 2, v.z);
  atomicAdd(dp + 3, v.w);
}

// ---------------- in-place relu(layernorm(x)) over CH=128, wave per row ----------------
__global__ void ln_relu_kernel(float* __restrict__ x, const float* __restrict__ g,
                               const float* __restrict__ b, int rows) {
  int wid = (int)((blockIdx.x * blockDim.x + threadIdx.x) >> 5);
  int lane = threadIdx.x & 31;
  if (wid >= rows) return;
  float* row = x + (size_t)wid * CH;
  float v0 = row[lane], v1 = row[lane + 32], v2 = row[lane + 64], v3 = row[lane + 96];
  float s = wave_sum(v0 + v1 + v2 + v3);
  float sq = wave_sum(v0 * v0 + v1 * v1 + v2 * v2 + v3 * v3);
  float mean = s * (1.0f / CH);
  float var = sq * (1.0f / CH) - mean * mean;
  float inv = rsqrtf(var + 1e-5f);
  row[lane]      = fmaxf((v0 - mean) * inv * g[lane]      + b[lane],      0.f);
  row[lane + 32] = fmaxf((v1 - mean) * inv * g[lane + 32] + b[lane + 32], 0.f);
  row[lane + 64] = fmaxf((v2 - mean) * inv * g[lane + 64] + b[lane + 64], 0.f);
  row[lane + 96] = fmaxf((v3 - mean) * inv * g[lane + 96] + b[lane + 96], 0.f);
}

// ---------------- y = l2norm(x) rowwise over CH=128, wave per row ----------------
__global__ void l2norm_rows_kernel(const float* __restrict__ x, float* __restrict__ y, int rows) {
  int wid = (int)((blockIdx.x * blockDim.x + threadIdx.x) >> 5);
  int lane = threadIdx.x & 31;
  if (wid >= rows) return;
  const float* row = x + (size_t)wid * CH;
  float v0 = row[lane], v1 = row[lane + 32], v2 = row[lane + 64], v3 = row[lane + 96];
  float s = wave_sum(v0 * v0 + v1 * v1 + v2 * v2 + v3 * v3);
  float sc = 1.0f / fmaxf(sqrtf(s), 1e-12f);
  float* o = y + (size_t)wid * CH;
  o[lane] = v0 * sc; o[lane + 32] = v1 * sc; o[lane + 64] = v2 * sc; o[lane + 96] = v3 * sc;
}

extern "C" void kernel_launch(void* const* d_in, const int* in_sizes, int n_in,
                              void* d_out, int out_size, void* d_ws, size_t ws_size,
                              hipStream_t stream) {
  const float* x_client = (const float*)d_in[0];
  const float* x_sku    = (const float*)d_in[1];
  const int* src_cs = (const int*)d_in[2];
  const int* dst_cs = (const int*)d_in[3];
  const int* src_sc = (const int*)d_in[4];
  const int* dst_sc = (const int*)d_in[5];
  const float* Wi_c = (const float*)d_in[6];
  const float* bi_c = (const float*)d_in[7];
  const float* Wi_s = (const float*)d_in[8];
  const float* bi_s = (const float*)d_in[9];
  const float* eps_gin = (const float*)d_in[10];  // [L,2]
  const float* mlp_W = (const float*)d_in[11];    // [L,2,2,CH,CH]
  const float* mlp_b = (const float*)d_in[12];    // [L,2,2,CH]
  const float* ln_g = (const float*)d_in[13];     // [L,2,CH]
  const float* ln_b = (const float*)d_in[14];
  const float* Wc1 = (const float*)d_in[15];   const float* bc1 = (const float*)d_in[16];
  const float* Wc2 = (const float*)d_in[17];   const float* bc2 = (const float*)d_in[18];
  const float* Wcat1 = (const float*)d_in[19]; const float* bcat1 = (const float*)d_in[20];
  const float* Wcat2 = (const float*)d_in[21]; const float* bcat2 = (const float*)d_in[22];
  const float* Wsku1 = (const float*)d_in[23]; const float* bsku1 = (const float*)d_in[24];
  const float* Wsku2 = (const float*)d_in[25]; const float* bsku2 = (const float*)d_in[26];

  // workspace carve-up (floats)
  float* ws = (float*)d_ws;
  size_t o = 0;
  float* xc   = ws + o; o += (size_t)NCLI * CH;
  float* xs   = ws + o; o += (size_t)NSKU_N * CH;
  float* aggC = ws + o; o += (size_t)NCLI * CH;
  float* aggS = ws + o; o += (size_t)NSKU_N * CH;
  float* tmp  = ws + o; o += (size_t)NCLI * CH;
  float* invc = ws + o; o += (size_t)NCLI;
  float* invs = ws + o; o += (size_t)NSKU_N;

  float* out = (float*)d_out;
  float* churn = out;                                            // [NC,1]
  float* cat   = out + (size_t)NCLI;                             // [NC,100]
  float* sku   = out + (size_t)NCLI * (1 + NCAT_O);              // [NC,2000]
  float* user  = out + (size_t)NCLI * (1 + NCAT_O + NSKU_O);     // [NC,128]

  auto gemm = [&](int act, const float* A, const float* rs, const float* W, const float* b,
                  float* Cp, int M, int N, int K) {
    dim3 grid(M / 16, (N + NBLK - 1) / NBLK);
    if (act == 0)
      gemm_wmma_kernel<0><<<grid, 256, 0, stream>>>(A, rs, W, b, Cp, M, N, K);
    else if (act == 1)
      gemm_wmma_kernel<1><<<grid, 256, 0, stream>>>(A, rs, W, b, Cp, M, N, K);
    else
      gemm_wmma_kernel<2><<<grid, 256, 0, stream>>>(A, rs, W, b, Cp, M, N, K);
  };

  // initial projections with fused l2norm row scaling
  rownorm_kernel<<<(NCLI + 7) / 8, 256, 0, stream>>>(x_client, invc, NCLI, FEAT);
  rownorm_kernel<<<(NSKU_N + 7) / 8, 256, 0, stream>>>(x_sku, invs, NSKU_N, FEAT);
  gemm(0, x_client, invc, Wi_c, bi_c, xc, NCLI, CH, FEAT);
  gemm(0, x_sku, invs, Wi_s, bi_s, xs, NSKU_N, CH, FEAT);

  const size_t nS = (size_t)NSKU_N * CH, nC = (size_t)NCLI * CH;
  for (int l = 0; l < 2; ++l) {
    // aggregate (both directions read pre-update xc/xs)
    scale_copy_kernel<<<(unsigned)((nS + 255) / 256), 256, 0, stream>>>(xs, aggS, eps_gin + (l * 2 + 0), nS);
    scatter_add_kernel<<<(NEDGE + 7) / 8, 256, 0, stream>>>(aggS, xc, src_cs, dst_cs, NEDGE);
    scale_copy_kernel<<<(unsigned)((nC + 255) / 256), 256, 0, stream>>>(xc, aggC, eps_gin + (l * 2 + 1), nC);
    scatter_add_kernel<<<(NEDGE + 7) / 8, 256, 0, stream>>>(aggC, xs, src_sc, dst_sc, NEDGE);

    const float* W00 = mlp_W + ((size_t)(l * 2 + 0) * 2 + 0) * CH * CH;
    const float* W01 = mlp_W + ((size_t)(l * 2 + 0) * 2 + 1) * CH * CH;
    const float* W10 = mlp_W + ((size_t)(l * 2 + 1) * 2 + 0) * CH * CH;
    const float* W11 = mlp_W + ((size_t)(l * 2 + 1) * 2 + 1) * CH * CH;
    const float* b00 = mlp_b + ((size_t)(l * 2 + 0) * 2 + 0) * CH;
    const float* b01 = mlp_b + ((size_t)(l * 2 + 0) * 2 + 1) * CH;
    const float* b10 = mlp_b + ((size_t)(l * 2 + 1) * 2 + 0) * CH;
    const float* b11 = mlp_b + ((size_t)(l * 2 + 1) * 2 + 1) * CH;

    // SKU update: xs = relu(LN(mlp(aggS)))
    gemm(1, aggS, nullptr, W00, b00, tmp, NSKU_N, CH, CH);
    gemm(1, tmp, nullptr, W01, b01, xs, NSKU_N, CH, CH);
    ln_relu_kernel<<<(NSKU_N + 7) / 8, 256, 0, stream>>>(xs, ln_g + (size_t)(l * 2 + 1) * CH,
                                                         ln_b + (size_t)(l * 2 + 1) * CH, NSKU_N);
    // client update: xc = relu(LN(mlp(aggC)))
    gemm(1, aggC, nullptr, W10, b10, tmp, NCLI, CH, CH);
    gemm(1, tmp, nullptr, W11, b11, xc, NCLI, CH, CH);
    ln_relu_kernel<<<(NCLI + 7) / 8, 256, 0, stream>>>(xc, ln_g + (size_t)(l * 2 + 0) * CH,
                                                       ln_b + (size_t)(l * 2 + 0) * CH, NCLI);
  }

  // user embedding + heads
  l2norm_rows_kernel<<<(NCLI + 7) / 8, 256, 0, stream>>>(xc, user, NCLI);
  gemm(1, user, nullptr, Wc1, bc1, tmp, NCLI, CH, CH);
  gemm(2, tmp, nullptr, Wc2, bc2, churn, NCLI, 1, CH);
  gemm(1, user, nullptr, Wcat1, bcat1, tmp, NCLI, CH, CH);
  gemm(2, tmp, nullptr, Wcat2, bcat2, cat, NCLI, NCAT_O, CH);
  gemm(1, user, nullptr, Wsku1, bsku1, tmp, NCLI, CH, CH);
  gemm(2, tmp, nullptr, Wsku2, bsku2, sku, NCLI, NSKU_O, CH);
}